// ReRepresentation_40931038331399
// MI455X (gfx1250) — compile-verified
//
#include <hip/hip_runtime.h>
#include <hip/hip_bf16.h>

typedef float v2f __attribute__((ext_vector_type(2)));
typedef float v8f __attribute__((ext_vector_type(8)));

constexpr int kB  = 4;
constexpr int kNS = 512;
constexpr int kNQ = 4096;
constexpr int kD  = 128;
constexpr int kNT = kNS + kNQ;          // 4608
constexpr float kTau = 0.1f;
constexpr float kA2  = 0.9f;
constexpr int NW  = 8;                  // waves per block
constexpr int QS  = 132;                // padded LDS row stride (floats), 16B-aligned rows
constexpr int PS  = 20;                 // P tile stride

// ---------------------------------------------------------------------------
// new_query == query (softmax row-sums are 1; einsum 'bij,bid->bid' collapses)
// so the query half of the output is a straight copy.
// ---------------------------------------------------------------------------
__global__ void copy_query_kernel(const float* __restrict__ feat,
                                  float* __restrict__ out) {
    const int perB = kNQ * kD / 4;                       // float4s per batch
    int i = blockIdx.x * blockDim.x + threadIdx.x;
    if (i >= kB * perB) return;
    int b = i / perB;
    int r = i - b * perB;
    size_t off = (size_t)b * kNT * (kD / 4) + (size_t)kNS * (kD / 4) + r;
    ((float4*)out)[off] = ((const float4*)feat)[off];
}

// ---------------------------------------------------------------------------
// Fused support->query distance attention (flash style, fp32 WMMA 16x16x4).
// One block = one batch x one 16-row support tile. 8 waves split the 4096
// queries; per-wave online softmax; LDS combine at the end.
// ---------------------------------------------------------------------------
__launch_bounds__(NW * 32, 1)
__global__ void support_attn_kernel(const float* __restrict__ feat,
                                    float* __restrict__ out) {
    __shared__ alignas(16) float sSup[16][QS];       // support tile 16x128
    __shared__ alignas(16) float sQ[NW][16][QS];     // per-wave query tile / acc dump
    __shared__ alignas(16) float sP[NW][16][PS];     // per-wave P transpose staging
    __shared__ float sNormS[16];
    __shared__ float sMw[NW][16];
    __shared__ float sSw[NW][16];

    const int tid  = threadIdx.x;
    const int wave = tid >> 5;
    const int lane = tid & 31;
    const int hf   = lane >> 4;          // half-wave (selects K pair / M group)
    const int ln   = lane & 15;

    const int b    = blockIdx.x >> 5;    // 32 tiles per batch
    const int tile = blockIdx.x & 31;
    const size_t batchBase = (size_t)b * kNT * kD;
    const float* supBase = feat + batchBase + (size_t)tile * 16 * kD;
    const float* qryBase = feat + batchBase + (size_t)kNS * kD;

    // ---- stage support tile + norms ----
    for (int i = tid; i < 16 * (kD / 4); i += NW * 32) {
        int row = i >> 5, c4 = i & 31;
        float4 v = ((const float4*)(supBase + (size_t)row * kD))[c4];
        *(float4*)&sSup[row][c4 * 4] = v;
    }
    __syncthreads();
    if (tid < 16) {
        float s = 0.f;
        for (int d = 0; d < kD; ++d) { float v = sSup[tid][d]; s += v * v; }
        sNormS[tid] = s;
    }
    __syncthreads();

    float snS[8];
    for (int r = 0; r < 8; ++r) snS[r] = sNormS[hf * 8 + r];

    v8f acc[8];
    for (int t = 0; t < 8; ++t)
        for (int r = 0; r < 8; ++r) acc[t][r] = 0.f;
    float Mrow[8], Srow[8];
    for (int r = 0; r < 8; ++r) { Mrow[r] = -1e30f; Srow[r] = 0.f; }

    // ---- query chunk loop (16 queries per chunk, NW-way interleaved) ----
    for (int qc = wave; qc < kNQ / 16; qc += NW) {
        const float* qt = qryBase + (size_t)qc * 16 * kD;
        for (int j = 0; j < 16; ++j) {
            float4 v = ((const float4*)(qt + (size_t)j * kD))[lane];
            *(float4*)&sQ[wave][j][lane * 4] = v;
            if (qc + NW < kNQ / 16)  // prefetch next chunk's row
                __builtin_prefetch(qt + (size_t)NW * 16 * kD + (size_t)j * kD + lane * 4, 0, 0);
        }
        asm volatile("" ::: "memory");   // DS ops are in-order per wave

        // per-lane query norm (lane's C-layout column n = ln)
        float snQ = 0.f;
        for (int c4 = 0; c4 < 32; ++c4) {
            float4 v = *(const float4*)&sQ[wave][ln][c4 * 4];
            snQ += v.x * v.x + v.y * v.y + v.z * v.z + v.w * v.w;
        }

        // ---- S = sup . qryT via 32 x wmma_f32_16x16x4 ----
        v8f c = {};
        for (int kk = 0; kk < 32; ++kk) {
            v2f a  = *(const v2f*)&sSup[ln][kk * 4 + hf * 2];
            v2f bb = *(const v2f*)&sQ[wave][ln][kk * 4 + hf * 2];
            c = __builtin_amdgcn_wmma_f32_16x16x4_f32(false, a, false, bb,
                                                      (short)0, c, false, false);
        }

        // ---- online softmax (row stats across 16 lanes of each half) ----
        float l[8], p[8], scale[8];
        for (int r = 0; r < 8; ++r)
            l[r] = -kTau * (snS[r] + snQ - 2.f * c[r]);
        for (int r = 0; r < 8; ++r) {
            float tmax = l[r];
            for (int off = 1; off < 16; off <<= 1)
                tmax = fmaxf(tmax, __shfl_xor(tmax, off, 16));
            float newM = fmaxf(Mrow[r], tmax);
            scale[r] = __expf(Mrow[r] - newM);
            p[r] = __expf(l[r] - newM);
            float rs = p[r];
            for (int off = 1; off < 16; off <<= 1)
                rs += __shfl_xor(rs, off, 16);
            Srow[r] = Srow[r] * scale[r] + rs;
            Mrow[r] = newM;
        }
        for (int t = 0; t < 8; ++t)
            for (int r = 0; r < 8; ++r) acc[t][r] *= scale[r];

        // ---- transpose P (C layout -> A layout) through LDS ----
        for (int r = 0; r < 8; ++r)
            sP[wave][hf * 8 + r][ln] = p[r];
        asm volatile("" ::: "memory");
        v2f pa[4];
        for (int kk = 0; kk < 4; ++kk)
            pa[kk] = *(const v2f*)&sP[wave][ln][kk * 4 + hf * 2];

        // ---- acc += P . V  (8 d-tiles x 4 k-steps) ----
        for (int t = 0; t < 8; ++t) {
            for (int kk = 0; kk < 4; ++kk) {
                v2f bb;
                bb.x = sQ[wave][kk * 4 + hf * 2 + 0][t * 16 + ln];
                bb.y = sQ[wave][kk * 4 + hf * 2 + 1][t * 16 + ln];
                acc[t] = __builtin_amdgcn_wmma_f32_16x16x4_f32(false, pa[kk], false, bb,
                                                               (short)0, acc[t], false, false);
            }
        }
        asm volatile("" ::: "memory");
    }

    // ---- cross-wave combine: dump partials into LDS (reuse sQ) ----
    for (int t = 0; t < 8; ++t)
        for (int r = 0; r < 8; ++r)
            sQ[wave][hf * 8 + r][t * 16 + ln] = acc[t][r];
    if (ln == 0)
        for (int r = 0; r < 8; ++r) {
            sMw[wave][hf * 8 + r] = Mrow[r];
            sSw[wave][hf * 8 + r] = Srow[r];
        }
    __syncthreads();

    float* outBase = out + batchBase + (size_t)tile * 16 * kD;
    for (int i = tid; i < 16 * kD; i += NW * 32) {
        int m = i >> 7, d = i & 127;
        float gM = -1e30f;
        for (int w = 0; w < NW; ++w) gM = fmaxf(gM, sMw[w][m]);
        float den = 0.f, num = 0.f;
        for (int w = 0; w < NW; ++w) {
            float f = __expf(sMw[w][m] - gM);
            den += sSw[w][m] * f;
            num += sQ[w][m][d] * f;
        }
        outBase[(size_t)m * kD + d] = (1.f - kA2) * sSup[m][d] + kA2 * (num / den);
    }
}

extern "C" void kernel_launch(void* const* d_in, const int* in_sizes, int n_in,
                              void* d_out, int out_size, void* d_ws, size_t ws_size,
                              hipStream_t stream) {
    const float* feat = (const float*)d_in[0];
    float* out = (float*)d_out;
    // query half: identity copy (see analysis)
    int nv4 = kB * kNQ * kD / 4;
    copy_query_kernel<<<(nv4 + 255) / 256, 256, 0, stream>>>(feat, out);
    // support half: fused distance attention
    support_attn_kernel<<<kB * (kNS / 16), NW * 32, 0, stream>>>(feat, out);
}